// GSN_matrix_8392366096422
// MI455X (gfx1250) — compile-verified
//
#include <hip/hip_runtime.h>
#include <math.h>

// GSN matrix update for MI455X (gfx1250, wave32, WMMA).
//
// Restructuring: the [B,N,N,D] GRU tensor has only 257 distinct input rows
// (hidden rows, with row 0 = zeros for adj==0 entries). We compute
// G_* = hidden[1:257] @ W_*^T once (4 fp32 WMMA GEMMs, 256x512x1536),
// a bias-only constant GRU output g0 per direction for adj==0 entries,
// then per output node sum the few real edge contributions elementwise.

typedef float v2f __attribute__((ext_vector_type(2)));
typedef float v8f __attribute__((ext_vector_type(8)));

#define DD   512
#define G3   1536
#define NROW 256
#define GSZ  ((size_t)NROW * G3)

__device__ __forceinline__ float sigmoidf_(float x) {
    return 1.0f / (1.0f + __expf(-x));
}

// ---------------------------------------------------------------------------
// Kernel 1: G[z] = hidden[1:257] @ W[z]^T   (M=256, K=512, N=1536), fp32 WMMA.
// Both operands are K-contiguous, so each lane's 16x16x4 fragment is one
// aligned float2 load (A: lane=M row, kb selects K half; B: lane=N row).
// ---------------------------------------------------------------------------
__global__ __launch_bounds__(128)
void gsn_gemm_wmma(const float* __restrict__ hidden,
                   const float* __restrict__ Wih_f, const float* __restrict__ Whh_f,
                   const float* __restrict__ Wih_b, const float* __restrict__ Whh_b,
                   float* __restrict__ ws)
{
    const int z = blockIdx.z;
    const float* W = (z == 0) ? Wih_f : (z == 1) ? Whh_f : (z == 2) ? Wih_b : Whh_b;
    float* G = ws + (size_t)z * GSZ;

    const int lane = threadIdx.x & 31;
    const int wave = threadIdx.x >> 5;
    const int n0 = (blockIdx.x * 4 + wave) * 16;
    const int m0 = blockIdx.y * 16;

    const int rc = lane & 15;            // A: M row; B: N row
    const int kb = (lane >> 4) << 1;     // K sub-offset: 0 (lanes 0-15) / 2 (16-31)

    const float* Aptr = hidden + DD /* skip zero row 0 */ + (size_t)(m0 + rc) * DD + kb;
    const float* Bptr = W + (size_t)(n0 + rc) * DD + kb;

    v8f c0 = {}, c1 = {}, c2 = {}, c3 = {};

    #pragma unroll 4
    for (int k = 0; k < DD; k += 16) {
        v2f a0 = *(const v2f*)(Aptr + k);
        v2f b0 = *(const v2f*)(Bptr + k);
        v2f a1 = *(const v2f*)(Aptr + k + 4);
        v2f b1 = *(const v2f*)(Bptr + k + 4);
        v2f a2 = *(const v2f*)(Aptr + k + 8);
        v2f b2 = *(const v2f*)(Bptr + k + 8);
        v2f a3 = *(const v2f*)(Aptr + k + 12);
        v2f b3 = *(const v2f*)(Bptr + k + 12);
        c0 = __builtin_amdgcn_wmma_f32_16x16x4_f32(false, a0, false, b0, (short)0, c0, false, false);
        c1 = __builtin_amdgcn_wmma_f32_16x16x4_f32(false, a1, false, b1, (short)0, c1, false, false);
        c2 = __builtin_amdgcn_wmma_f32_16x16x4_f32(false, a2, false, b2, (short)0, c2, false, false);
        c3 = __builtin_amdgcn_wmma_f32_16x16x4_f32(false, a3, false, b3, (short)0, c3, false, false);
    }
    v8f d = (c0 + c1) + (c2 + c3);

    const int mhalf = (lane >> 4) * 8;   // C/D layout: lanes 16-31 hold M=v+8
    const int col = n0 + rc;
    #pragma unroll
    for (int v = 0; v < 8; ++v) {
        G[(size_t)(m0 + mhalf + v) * G3 + col] = d[v];
    }
}

// ---------------------------------------------------------------------------
// Kernel 2: bias-only GRU outputs g0_f / g0_b (for adj==0 entries), and
// output row 0 passthrough (hidden row 0 == zeros).
// ---------------------------------------------------------------------------
__global__ __launch_bounds__(512)
void gsn_init(const float* __restrict__ hidden,
              const float* __restrict__ bih_f, const float* __restrict__ bhh_f,
              const float* __restrict__ bih_b, const float* __restrict__ bhh_b,
              float* __restrict__ g0f, float* __restrict__ g0b,
              float* __restrict__ out)
{
    const int d = threadIdx.x;  // 0..511
    {
        float r  = sigmoidf_(bih_f[d]        + bhh_f[d]);
        float zz = sigmoidf_(bih_f[DD + d]   + bhh_f[DD + d]);
        float n  = tanhf(bih_f[2*DD + d] + r * bhh_f[2*DD + d]);
        g0f[d] = (1.0f - zz) * n;     // h == 0 for the zero row
    }
    {
        float r  = sigmoidf_(bih_b[d]        + bhh_b[d]);
        float zz = sigmoidf_(bih_b[DD + d]   + bhh_b[DD + d]);
        float n  = tanhf(bih_b[2*DD + d] + r * bhh_b[2*DD + d]);
        g0b[d] = (1.0f - zz) * n;
    }
    out[d] = hidden[d];
}

// ---------------------------------------------------------------------------
// Kernel 3: one block per output node pj = b*32 + j. Sum over i of:
//   adj[b,i,j]==1 ? gru_b(x=child_i, h=parent_j) + gru_f(x=parent_j, h=child_i)
//                 : g0_b + g0_f
// plus the hidden passthrough. Node-j-invariant gate terms are hoisted.
// ---------------------------------------------------------------------------
__global__ __launch_bounds__(256)
void gsn_edges(const float* __restrict__ hidden,
               const int* __restrict__ adj,
               const float* __restrict__ ws,
               const float* __restrict__ bih_f, const float* __restrict__ bhh_f,
               const float* __restrict__ bih_b, const float* __restrict__ bhh_b,
               float* __restrict__ out)
{
    const float* Gih_f = ws;
    const float* Ghh_f = ws + GSZ;
    const float* Gih_b = ws + 2 * GSZ;
    const float* Ghh_b = ws + 3 * GSZ;
    const float* g0f   = ws + 4 * GSZ;
    const float* g0b   = g0f + DD;

    const int node = blockIdx.x;   // pj in [0,256)
    const int b = node >> 5;
    const int j = node & 31;

    __shared__ int adjcol[32];
    if (threadIdx.x < 32) adjcol[threadIdx.x] = adj[((b * 32 + threadIdx.x) * 32) + j];
    __syncthreads();

    for (int d = (int)threadIdx.x; d < DD; d += 256) {
        // node-j invariant gate inputs (biases folded in)
        const float* gj_if = Gih_f + (size_t)node * G3;
        const float* gj_hb = Ghh_b + (size_t)node * G3;
        const float xf_r = gj_if[d]          + bih_f[d];
        const float xf_z = gj_if[DD + d]     + bih_f[DD + d];
        const float xf_n = gj_if[2*DD + d]   + bih_f[2*DD + d];
        const float hb_r = gj_hb[d]          + bhh_b[d];
        const float hb_z = gj_hb[DD + d]     + bhh_b[DD + d];
        const float hb_n = gj_hb[2*DD + d]   + bhh_b[2*DD + d];
        const float h_j  = hidden[(size_t)(node + 1) * DD + d];

        const float bibr = bih_b[d], bibz = bih_b[DD + d], bibn = bih_b[2*DD + d];
        const float bhfr = bhh_f[d], bhfz = bhh_f[DD + d], bhfn = bhh_f[2*DD + d];

        float acc = h_j;
        int zeros = 0;
        for (int i = 0; i < 32; ++i) {
            if (adjcol[i] == 0) { zeros++; continue; }
            const int ci = b * 32 + i;
            const float* gxb = Gih_b + (size_t)ci * G3;
            const float* ghf = Ghh_f + (size_t)ci * G3;
            const float h_i = hidden[(size_t)(ci + 1) * DD + d];
            // backward GRU: input = child i, hidden = parent j  -> p_change
            {
                float r  = sigmoidf_((gxb[d] + bibr)        + hb_r);
                float zz = sigmoidf_((gxb[DD + d] + bibz)   + hb_z);
                float n  = tanhf((gxb[2*DD + d] + bibn) + r * hb_n);
                acc += (1.0f - zz) * n + zz * h_j;
            }
            // forward GRU: input = parent j, hidden = child i -> c_change
            {
                float r  = sigmoidf_(xf_r + (ghf[d] + bhfr));
                float zz = sigmoidf_(xf_z + (ghf[DD + d] + bhfz));
                float n  = tanhf(xf_n + r * (ghf[2*DD + d] + bhfn));
                acc += (1.0f - zz) * n + zz * h_i;
            }
        }
        acc += (float)zeros * (g0f[d] + g0b[d]);
        out[(size_t)(node + 1) * DD + d] = acc;
    }
}

extern "C" void kernel_launch(void* const* d_in, const int* in_sizes, int n_in,
                              void* d_out, int out_size, void* d_ws, size_t ws_size,
                              hipStream_t stream) {
    const float* hidden = (const float*)d_in[0];
    // d_in[1] (state_matrix) is structurally determined -> unused
    const int*   adj    = (const int*)d_in[2];
    const float* Wih_f  = (const float*)d_in[3];
    const float* Whh_f  = (const float*)d_in[4];
    const float* bih_f  = (const float*)d_in[5];
    const float* bhh_f  = (const float*)d_in[6];
    const float* Wih_b  = (const float*)d_in[7];
    const float* Whh_b  = (const float*)d_in[8];
    const float* bih_b  = (const float*)d_in[9];
    const float* bhh_b  = (const float*)d_in[10];
    float* out = (float*)d_out;
    float* ws  = (float*)d_ws;   // needs 4*256*1536*4 + 2*512*4 bytes ~= 6.3 MB

    // 1) G matrices: grid (1536/64, 256/16, 4 weight matrices), 4 waves/block
    gsn_gemm_wmma<<<dim3(24, 16, 4), 128, 0, stream>>>(hidden, Wih_f, Whh_f,
                                                       Wih_b, Whh_b, ws);
    // 2) bias-only GRU constants + out row 0
    gsn_init<<<1, 512, 0, stream>>>(hidden, bih_f, bhh_f, bih_b, bhh_b,
                                    ws + 4 * GSZ, ws + 4 * GSZ + DD, out);
    // 3) per-node edge accumulation
    gsn_edges<<<256, 256, 0, stream>>>(hidden, adj, ws,
                                       bih_f, bhh_f, bih_b, bhh_b, out);
}